// Embedding_5557687681688
// MI455X (gfx1250) — compile-verified
//
#include <hip/hip_runtime.h>

// CDNA5 / gfx1250, wave32.
typedef __attribute__((ext_vector_type(2))) float v2f;
typedef __attribute__((ext_vector_type(8))) float v8f;

// out[b,i,n] = relu(x[b,i]*w1[i]+b1[i]) * W2[i,n] + b2[i,n]
// Rank-2 trick: D = A(16x4) x B(4x16) with A=[h | 1 | 0 | 0], B=[W2row ; b2row ; 0 ; 0]
__global__ __launch_bounds__(256) void mlp_outer_wmma(
    const float* __restrict__ x,
    const float* __restrict__ w1,
    const float* __restrict__ b1,
    const float* __restrict__ W2,
    const float* __restrict__ b2,
    float* __restrict__ out,
    int Btiles) {
  constexpr int V = 64;
  constexpr int N = 256;

  const unsigned lane = threadIdx.x & 31u;           // wave32 lane
  const unsigned waveInBlock = threadIdx.x >> 5;     // 0..7
  const unsigned wave = blockIdx.x * (blockDim.x >> 5) + waveInBlock;

  const unsigned i  = wave & (V - 1);                // dimension index 0..63
  const unsigned bt = wave >> 6;                     // b-tile index 0..255
  if (bt >= (unsigned)Btiles) return;                // wave-uniform guard
  const unsigned b0 = bt * 16u;

  const unsigned m  = lane & 15u;                    // row/col within tile
  const bool lo = lane < 16u;                        // lanes 0-15 carry K0/K1

  // h[b0+m, i] = relu(x*w1+b1); all 32 lanes read a valid (duplicated) address.
  const float w1i = w1[i];
  const float b1i = b1[i];
  const float xv  = x[(size_t)(b0 + m) * V + i];
  const float h   = fmaxf(fmaf(xv, w1i, b1i), 0.0f);

  // A-matrix (16x4 f32, 2 VGPRs): lanes 0-15 -> {K0=h, K1=1}; lanes 16-31 -> {0,0}
  v2f a;
  a.x = lo ? h    : 0.0f;
  a.y = lo ? 1.0f : 0.0f;

  const float* __restrict__ w2row = W2 + (size_t)i * N;
  const float* __restrict__ b2row = b2 + (size_t)i * N;
  float* __restrict__ orow = out + ((size_t)b0 * V + i) * N;
  const size_t rstride = (size_t)V * N;              // elements between b-rows
  const unsigned rbase = lo ? 0u : 8u;               // D layout: M = r (+8 for hi half)

#pragma unroll 4
  for (int n0 = 0; n0 < N; n0 += 16) {
    // B-matrix (4x16 f32, 2 VGPRs): lanes 0-15 -> {K0=W2, K1=b2}; lanes 16-31 -> {0,0}
    const float wv = w2row[n0 + m];
    const float bv = b2row[n0 + m];
    v2f bb;
    bb.x = lo ? wv : 0.0f;
    bb.y = lo ? bv : 0.0f;

    v8f c = {};
    // (neg_a, A, neg_b, B, c_mod, C, reuse_a, reuse_b)
    c = __builtin_amdgcn_wmma_f32_16x16x4_f32(
        /*neg_a=*/false, a, /*neg_b=*/false, bb,
        /*c_mod=*/(short)0, c, /*reuse_a=*/false, /*reuse_b=*/false);

    // D tile store: VGPR r, lanes 0-15 -> (b0+r, n0+lane); lanes 16-31 -> (b0+r+8, n0+lane-16)
    float* __restrict__ tp = orow + n0 + m;
#pragma unroll
    for (int r = 0; r < 8; ++r) {
      __builtin_nontemporal_store(c[r], tp + (size_t)(rbase + r) * rstride);
    }
  }
}

extern "C" void kernel_launch(void* const* d_in, const int* in_sizes, int n_in,
                              void* d_out, int out_size, void* d_ws, size_t ws_size,
                              hipStream_t stream) {
  const float* x  = (const float*)d_in[0];
  const float* w1 = (const float*)d_in[1];
  const float* b1 = (const float*)d_in[2];
  const float* W2 = (const float*)d_in[3];
  const float* b2 = (const float*)d_in[4];
  float* out = (float*)d_out;

  const int V = 64;
  const int B = in_sizes[0] / V;        // 4096
  const int Btiles = B / 16;            // 256
  const int totalWaves = Btiles * V;    // 16384 waves, one 16x256 output slab each
  const int wavesPerBlock = 8;          // 256 threads = 8 wave32 waves
  const int blocks = (totalWaves + wavesPerBlock - 1) / wavesPerBlock;

  mlp_outer_wmma<<<blocks, wavesPerBlock * 32, 0, stream>>>(
      x, w1, b1, W2, b2, out, Btiles);
}